// SteadyIncompressiblePINN_35485019799661
// MI455X (gfx1250) — compile-verified
//
#include <hip/hip_runtime.h>
#include <cstdint>

#define NPTS 16384
#define MQ   4096
#define NSPLIT 8                  // waves per query tile (candidate-dim split)
#define CHUNK (NPTS / NSPLIT)     // 2048 candidates per wave
#define EPSF 1e-8f
#define RIDGEF 1e-6f
#define NUF 0.01f
#define SIGMAF 0.08f

typedef __attribute__((ext_vector_type(2))) float v2f;
typedef __attribute__((ext_vector_type(8))) float v8f;

// ---------------------------------------------------------------------------
// K0: build A-matrix rows axz[n] = (-2x,-2y,-2z,|x|^2) and query tile
//     q4[m] = (qx,qy,qz,1).  Score = axz . q4 = |x|^2 - 2 x.q  (== d^2 - |q|^2,
//     rank-equivalent to d^2 per query).
// ---------------------------------------------------------------------------
__global__ void prep_kernel(const float* __restrict__ xyz,
                            const int* __restrict__ sidx,
                            float* __restrict__ q4,
                            float* __restrict__ axz) {
  int t = blockIdx.x * blockDim.x + threadIdx.x;
  if (t < NPTS) {
    float x = xyz[3 * t], y = xyz[3 * t + 1], z = xyz[3 * t + 2];
    *(float4*)(axz + 4 * t) =
        make_float4(-2.f * x, -2.f * y, -2.f * z, x * x + y * y + z * z);
  }
  if (t < MQ) {
    int i = sidx[t];
    *(float4*)(q4 + 4 * t) =
        make_float4(xyz[3 * i], xyz[3 * i + 1], xyz[3 * i + 2], 1.f);
  }
}

// ---------------------------------------------------------------------------
// K1: WMMA top-16 KNN, candidate dimension split across 8 waves per block.
//   Block (256 thr, 8 waves) owns one 16-query tile; wave w scans candidates
//   [w*2048, (w+1)*2048) with one V_WMMA_F32_16X16X4_F32 per 16 candidates:
//   A = 16 candidates x (-2x,-2y,-2z,|x|^2), B = (q;1) for the 16 queries.
//   Lane j / j+16 own rows 0-7 / 8-15 of query j.  Per-lane register top-16
//   with replace-max + lazy rescan, guarded by a per-WMMA min8 test.
//   Pipelined A-tile load + prefetch are branch-free (clamped addresses).
//   Merge: each wave min-extracts its 32-entry per-query list into a sorted
//   16-list (parallel over 8 waves), then wave 0 does an 8-way sorted merge.
// ---------------------------------------------------------------------------
__global__ void __launch_bounds__(256)
knn_kernel(const float* __restrict__ axz,
           const float* __restrict__ q4,
           int* __restrict__ knn_out, int* __restrict__ nn_out) {
  __shared__ float raw_s[NSPLIT][16][32];
  __shared__ int   raw_i[NSPLIT][16][32];
  __shared__ float srt_s[NSPLIT][16][17];   // +1 pad = INF sentinel
  __shared__ int   srt_i[NSPLIT][16][17];

  const int lane = threadIdx.x & 31;
  const int wave = threadIdx.x >> 5;
  const int qbase = blockIdx.x * 16;
  const int col = lane & 15;
  const int hi  = lane >> 4;      // lanes 16-31 -> K pair 2,3 / rows 8-15

  // B operand: 4x16, column = query. Low lanes hold (K0,K1)=(qx,qy),
  // high lanes hold (K2,K3)=(qz,1).
  const float4 qv = *(const float4*)(q4 + 4 * (qbase + col));
  v2f bvec;
  bvec.x = hi ? qv.z : qv.x;
  bvec.y = hi ? qv.w : qv.y;

  float s[16]; int id[16];
#pragma unroll
  for (int t = 0; t < 16; ++t) { s[t] = __builtin_inff(); id[t] = 0; }
  float wmax = __builtin_inff(); int pmax = 0;

  const int ksel = hi * 2;        // which K pair of the A row this lane supplies
  const int cbeg = wave * CHUNK;
  const int cend = cbeg + CHUNK;
  const int clast = cend - 16;    // last valid tile base (for clamped accesses)

  v2f a = *(const v2f*)(axz + 4 * (cbeg + col) + ksel);
  for (int base = cbeg; base < cend; base += 16) {
    // branch-free software pipeline: clamp next-tile / prefetch addresses
    const int nb = (base + 16 < clast) ? base + 16 : clast;
    const int pf = (base + 128 < clast) ? base + 128 : clast;
    v2f a_next = *(const v2f*)(axz + 4 * (nb + col) + ksel);
    __builtin_prefetch(axz + 4 * (pf + col), 0, 3);  // global_prefetch_b8, near scope

    v8f c = {0.f, 0.f, 0.f, 0.f, 0.f, 0.f, 0.f, 0.f};
    v8f d = __builtin_amdgcn_wmma_f32_16x16x4_f32(
        false, a, false, bvec, (short)0, c, false, false);

    // one cheap guard per WMMA: skip all inserts unless something can beat wmax
    float m8 = fminf(fminf(fminf(d[0], d[1]), fminf(d[2], d[3])),
                     fminf(fminf(d[4], d[5]), fminf(d[6], d[7])));
    if (m8 < wmax) {
#pragma unroll
      for (int r = 0; r < 8; ++r) {
        float sc = d[r];
        int cand = base + (hi << 3) + r;
        if (sc < wmax) {
#pragma unroll
          for (int t = 0; t < 16; ++t)
            if (t == pmax) { s[t] = sc; id[t] = cand; }
          float m = s[0]; int p = 0;
#pragma unroll
          for (int t = 1; t < 16; ++t)
            if (s[t] > m) { m = s[t]; p = t; }
          wmax = m; pmax = p;
        }
      }
    }
    a = a_next;
  }

  // dump the two 16-entry half-lists per (wave, query) to LDS
#pragma unroll
  for (int t = 0; t < 16; ++t) {
    raw_s[wave][col][hi * 16 + t] = s[t];
    raw_i[wave][col][hi * 16 + t] = id[t];
  }
  __syncthreads();

  // stage 1: each wave sorts its own 32-entry list per query (16 extractions),
  // parallel over the 8 waves; lane j handles query j.
  if (lane < 16) {
#pragma unroll 1
    for (int k = 0; k < 16; ++k) {
      float m = __builtin_inff(); int p = 0;
      for (int t = 0; t < 32; ++t) {
        float v = raw_s[wave][lane][t];
        if (v < m) { m = v; p = t; }
      }
      srt_s[wave][lane][k] = m;
      srt_i[wave][lane][k] = raw_i[wave][lane][p];
      raw_s[wave][lane][p] = __builtin_inff();
    }
    srt_s[wave][lane][16] = __builtin_inff();  // sentinel
    srt_i[wave][lane][16] = 0;
  }
  __syncthreads();

  // stage 2: wave 0, lane j: 8-way merge of sorted lists for query j.
  if (wave == 0 && lane < 16) {
    const int q = qbase + lane;
    int ptr[NSPLIT];
    float head[NSPLIT];
#pragma unroll
    for (int w = 0; w < NSPLIT; ++w) { ptr[w] = 0; head[w] = srt_s[w][lane][0]; }
#pragma unroll 1
    for (int k = 0; k < 16; ++k) {
      float m = head[0]; int wsel = 0;
#pragma unroll
      for (int w = 1; w < NSPLIT; ++w)
        if (head[w] < m) { m = head[w]; wsel = w; }
      int best = 0;
#pragma unroll
      for (int w = 0; w < NSPLIT; ++w)
        if (w == wsel) {
          best = srt_i[w][lane][ptr[w]];
          ptr[w] = ptr[w] + 1;
          head[w] = srt_s[w][lane][ptr[w]];
        }
      knn_out[q * 16 + k] = best;
      if (k == 0) nn_out[q] = best;   // global argmin == first merged element
    }
  }
}

// ---------------------------------------------------------------------------
// K2: per-query physics. IDW Laplacian with 0.99-quantile weight clip
// (interp of the two largest of 16), symmetric 3x3 ridge LSQ via adjugate,
// momentum residual, wall weight. Emits r and r*(div^2+|mom|^2).
// ---------------------------------------------------------------------------
__global__ void physics_kernel(const float* __restrict__ xyz,
                               const float* __restrict__ pred,
                               const float* __restrict__ q4,
                               const int* __restrict__ knn_in,
                               const int* __restrict__ nn_in,
                               float* __restrict__ re_out,
                               float* __restrict__ r_out) {
  int m = blockIdx.x * blockDim.x + threadIdx.x;
  if (m >= MQ) return;
  const float qx = q4[4 * m], qy = q4[4 * m + 1], qz = q4[4 * m + 2];
  const float4 fq = *(const float4*)(pred + 4 * nn_in[m]);

  // pass 1: weights and the two largest (for the 0.99 quantile)
  float w[16];
  float w1 = -__builtin_inff(), w2 = -__builtin_inff();
#pragma unroll
  for (int k = 0; k < 16; ++k) {
    int i = knn_in[16 * m + k];
    float dx = xyz[3 * i] - qx, dy = xyz[3 * i + 1] - qy, dz = xyz[3 * i + 2] - qz;
    float wk = 1.f / (dx * dx + dy * dy + dz * dz + EPSF);
    w[k] = wk;
    if (wk > w1) { w2 = w1; w1 = wk; }
    else if (wk > w2) { w2 = wk; }
  }
  const float thr = w2 + 0.85f * (w1 - w2);   // quantile(w,0.99) of 16 samples

  // pass 2: clipped-IDW numerator + XTX / XTy accumulation
  float sw = 0.f, num1 = 0.f, num2 = 0.f, num3 = 0.f;
  float xx = 0, xy = 0, xz = 0, yy = 0, yz = 0, zz = 0;
  float t00 = 0, t01 = 0, t02 = 0, t03 = 0;
  float t10 = 0, t11 = 0, t12 = 0, t13 = 0;
  float t20 = 0, t21 = 0, t22 = 0, t23 = 0;
#pragma unroll
  for (int k = 0; k < 16; ++k) {
    int i = knn_in[16 * m + k];
    float dx = xyz[3 * i] - qx, dy = xyz[3 * i + 1] - qy, dz = xyz[3 * i + 2] - qz;
    float4 fn = *(const float4*)(pred + 4 * i);
    float df0 = fn.x - fq.x, df1 = fn.y - fq.y, df2 = fn.z - fq.z, df3 = fn.w - fq.w;
    float wc = fminf(w[k], thr);
    sw += wc; num1 += wc * df1; num2 += wc * df2; num3 += wc * df3;
    xx += dx * dx; xy += dx * dy; xz += dx * dz;
    yy += dy * dy; yz += dy * dz; zz += dz * dz;
    t00 += dx * df0; t01 += dx * df1; t02 += dx * df2; t03 += dx * df3;
    t10 += dy * df0; t11 += dy * df1; t12 += dy * df2; t13 += dy * df3;
    t20 += dz * df0; t21 += dz * df1; t22 += dz * df2; t23 += dz * df3;
  }
  const float invsw = 1.f / fmaxf(sw, EPSF);
  const float lap1 = num1 * invsw, lap2 = num2 * invsw, lap3 = num3 * invsw;

  const float basem = (fabsf(xx) + fabsf(yy) + fabsf(zz) +
                       2.f * (fabsf(xy) + fabsf(xz) + fabsf(yz))) * (1.f / 9.f)
                      + 1e-12f;
  const float rg = RIDGEF * basem;
  const float a00 = xx + rg, a11 = yy + rg, a22 = zz + rg;
  const float a01 = xy, a02 = xz, a12 = yz;
  const float c00 = a11 * a22 - a12 * a12;
  const float c01 = a02 * a12 - a01 * a22;
  const float c02 = a01 * a12 - a02 * a11;
  const float det = a00 * c00 + a01 * c01 + a02 * c02;
  const float invd = 1.f / det;
  const float i00 = c00 * invd, i01 = c01 * invd, i02 = c02 * invd;
  const float i11 = (a00 * a22 - a02 * a02) * invd;
  const float i12 = (a01 * a02 - a00 * a12) * invd;
  const float i22 = (a00 * a11 - a01 * a01) * invd;

  // G[i][c] = (A^-1 XTy)[i][c]
  const float G00 = i00 * t00 + i01 * t10 + i02 * t20;
  const float G01 = i00 * t01 + i01 * t11 + i02 * t21;
  const float G02 = i00 * t02 + i01 * t12 + i02 * t22;
  const float G03 = i00 * t03 + i01 * t13 + i02 * t23;
  const float G10 = i01 * t00 + i11 * t10 + i12 * t20;
  const float G11 = i01 * t01 + i11 * t11 + i12 * t21;
  const float G12 = i01 * t02 + i11 * t12 + i12 * t22;
  const float G13 = i01 * t03 + i11 * t13 + i12 * t23;
  const float G20 = i02 * t00 + i12 * t10 + i22 * t20;
  const float G21 = i02 * t01 + i12 * t11 + i22 * t21;
  const float G22 = i02 * t02 + i12 * t12 + i22 * t22;
  const float G23 = i02 * t03 + i12 * t13 + i22 * t23;

  const float div = G01 + G12 + G23;
  const float u0 = fq.y, u1 = fq.z, u2 = fq.w;
  const float mom0 = (u0 * G01 + u1 * G11 + u2 * G21) + G00 - NUF * lap1;
  const float mom1 = (u0 * G02 + u1 * G12 + u2 * G22) + G10 - NUF * lap2;
  const float mom2 = (u0 * G03 + u1 * G13 + u2 * G23) + G20 - NUF * lap3;

  float dwall = fminf(fminf(fminf(qx, 1.f - qx), fminf(qy, 1.f - qy)),
                      fminf(qz, 1.f - qz));
  const float r = 1.f + __expf(-dwall / SIGMAF);
  const float e = div * div + mom0 * mom0 + mom1 * mom1 + mom2 * mom2;
  re_out[m] = r * e;
  r_out[m] = r;
}

// ---------------------------------------------------------------------------
// K3: deterministic single-block reduction.
// total = sum(r*e) / max(sum(r), M*EPS)   (wres normalization folded in)
// ---------------------------------------------------------------------------
__global__ void reduce_kernel(const float* __restrict__ re,
                              const float* __restrict__ r,
                              float* __restrict__ out) {
  __shared__ float s0[256], s1[256];
  float a = 0.f, b = 0.f;
  for (int i = threadIdx.x; i < MQ; i += 256) { a += re[i]; b += r[i]; }
  s0[threadIdx.x] = a; s1[threadIdx.x] = b;
  __syncthreads();
  for (int off = 128; off > 0; off >>= 1) {
    if (threadIdx.x < off) {
      s0[threadIdx.x] += s0[threadIdx.x + off];
      s1[threadIdx.x] += s1[threadIdx.x + off];
    }
    __syncthreads();
  }
  if (threadIdx.x == 0)
    out[0] = s0[0] / fmaxf(s1[0], (float)MQ * EPSF);
}

// ---------------------------------------------------------------------------
extern "C" void kernel_launch(void* const* d_in, const int* in_sizes, int n_in,
                              void* d_out, int out_size, void* d_ws, size_t ws_size,
                              hipStream_t stream) {
  (void)in_sizes; (void)n_in; (void)out_size; (void)ws_size;
  const float* xyz  = (const float*)d_in[0];   // (1, N, 3)
  const float* pred = (const float*)d_in[1];   // (1, N, 4)
  const int*   sidx = (const int*)d_in[2];     // (M,)
  float* out = (float*)d_out;

  char* ws = (char*)d_ws;
  float* q4  = (float*)ws;  ws += (size_t)MQ * 4 * sizeof(float);
  float* axz = (float*)ws;  ws += (size_t)NPTS * 4 * sizeof(float);
  int*   knn = (int*)ws;    ws += (size_t)MQ * 16 * sizeof(int);
  int*   nn  = (int*)ws;    ws += (size_t)MQ * sizeof(int);
  float* re  = (float*)ws;  ws += (size_t)MQ * sizeof(float);
  float* rr  = (float*)ws;

  prep_kernel<<<(NPTS + 255) / 256, 256, 0, stream>>>(xyz, sidx, q4, axz);
  knn_kernel<<<MQ / 16, 256, 0, stream>>>(axz, q4, knn, nn);
  physics_kernel<<<(MQ + 255) / 256, 256, 0, stream>>>(xyz, pred, q4, knn, nn, re, rr);
  reduce_kernel<<<1, 256, 0, stream>>>(re, rr, out);
}